// APPNProp_3178275799597
// MI455X (gfx1250) — compile-verified
//
#include <hip/hip_runtime.h>

// ---------------- problem constants (from reference) ----------------
#define D_FEAT 64
#define ALPHA 0.1f
#define K_STEPS 10

// ---------------- scatter kernel tiling ----------------
#define TILE 256            // edges staged per block-tile (== blockDim.x)
#define TILES_PER_BLOCK 10  // 2560 edges per block; 3.2M / 2560 = 1250 blocks exact
#define EPB (TILE * TILES_PER_BLOCK)

// Low 32 bits of a generic pointer to __shared__ == LDS byte offset
// (flat LDS addresses are {aperture_hi32, lds_offset_lo32}).
__device__ __forceinline__ unsigned lds_off_u32(const void* p) {
    return (unsigned)(unsigned long long)p;
}

// CDNA5 async global->LDS copy, GVS mode: mem = SGPR64 + VGPR32 offset.
// LDS dest byte address comes from a per-lane VGPR. Tracked by ASYNCcnt.
__device__ __forceinline__ void async_cp_b32(unsigned lds_byte_off,
                                             unsigned gbl_byte_off,
                                             const void* sgpr_base) {
    asm volatile("global_load_async_to_lds_b32 %0, %1, %2"
                 :
                 : "v"(lds_byte_off), "v"(gbl_byte_off), "s"(sgpr_base)
                 : "memory");
}

// Native non-returning FP32 atomic adds in GVS addressing form:
//   mem = SGPR64(base) + VGPR32(byte_off) + imm_offset
// Uniform base stays in SGPRs; one 32-bit per-lane offset serves both
// feature elements (second uses offset:4). STOREcnt-tracked, no return
// traffic, no CAS loop, no 64-bit VGPR address arithmetic.
__device__ __forceinline__ void atomic_add_f32x2_gvs(const float* sgpr_base,
                                                     unsigned byte_off,
                                                     float v0, float v1) {
    asm volatile("global_atomic_add_f32 %0, %1, %3\n\t"
                 "global_atomic_add_f32 %0, %2, %3 offset:4"
                 :
                 : "v"(byte_off), "v"(v0), "v"(v1), "s"(sgpr_base)
                 : "memory");
}

__global__ void __launch_bounds__(256)
appnp_scatter(const float* __restrict__ hsrc,
              const int*   __restrict__ erow,
              const int*   __restrict__ ecol,
              const float* __restrict__ ew,
              float*       __restrict__ agg,
              int E)
{
    // double-buffered edge staging; wave w only ever touches slots [w*32, w*32+32)
    __shared__ int   s_row[2][TILE];
    __shared__ int   s_col[2][TILE];
    __shared__ float s_w  [2][TILE];

    const int tid  = threadIdx.x;
    const int lane = tid & 31;
    const int wv   = tid >> 5;
    const int blockEdge0 = blockIdx.x * EPB;

    // Issue one tile's edge metadata (3 async ops per wave, EXEC always full:
    // OOB lanes are clamped so ASYNCcnt bookkeeping stays uniform).
    auto issue = [&](int t, int b) {
        int idx = blockEdge0 + t * TILE + tid;
        idx = (idx < E) ? idx : (E - 1);   // harmless duplicate load, guarded at use
        const unsigned go = (unsigned)idx << 2;
        async_cp_b32(lds_off_u32(&s_row[b][tid]), go, erow);
        async_cp_b32(lds_off_u32(&s_col[b][tid]), go, ecol);
        async_cp_b32(lds_off_u32(&s_w  [b][tid]), go, ew);
    };

    issue(0, 0);

    for (int t = 0; t < TILES_PER_BLOCK; ++t) {
        const int b = t & 1;
        if (t + 1 < TILES_PER_BLOCK) {
            issue(t + 1, b ^ 1);
            // 3 newly-issued ops may remain outstanding; the previous tile's
            // 3 ops (in-order per wave) must have retired.
            asm volatile("s_wait_asynccnt 0x3" ::: "memory");
        } else {
            asm volatile("s_wait_asynccnt 0x0" ::: "memory");
        }

        const int ebase = blockEdge0 + t * TILE + wv * 32;
        const int sbase = wv * 32;
#pragma unroll 4
        for (int e = 0; e < 32; ++e) {
            const int idx = ebase + e;
            if (idx < E) {
                // broadcast LDS reads (all 32 lanes, same address)
                const int   r = s_row[b][sbase + e];
                const int   c = s_col[b][sbase + e];
                const float w = s_w  [b][sbase + e];

                // prefetch next edge's feature row (global_prefetch_b8)
                if (e + 1 < 32) {
                    const int cn = s_col[b][sbase + e + 1];
                    __builtin_prefetch(hsrc + (size_t)cn * D_FEAT + lane * 2, 0, 0);
                }

                // coalesced 256B gather: 32 lanes x float2
                const float2 hv =
                    *(const float2*)(hsrc + (size_t)c * D_FEAT + lane * 2);

                // scatter: base + (r*256 + lane*8) bytes, elements at +0 / +4
                const unsigned dstOff =
                    (unsigned)r * (D_FEAT * 4u) + (unsigned)lane * 8u;
                atomic_add_f32x2_gvs(agg, dstOff, w * hv.x, w * hv.y);
            }
        }
    }
}

// h = (1-alpha)*agg + alpha*x, and re-zero agg for the next iteration.
__global__ void __launch_bounds__(256)
appnp_combine(const float4* __restrict__ x4,
              float4*       __restrict__ agg4,
              float4*       __restrict__ out4,
              int n4)
{
    const int i = blockIdx.x * blockDim.x + threadIdx.x;
    if (i < n4) {
        const float4 a  = agg4[i];
        const float4 xv = x4[i];
        float4 r;
        r.x = (1.0f - ALPHA) * a.x + ALPHA * xv.x;
        r.y = (1.0f - ALPHA) * a.y + ALPHA * xv.y;
        r.z = (1.0f - ALPHA) * a.z + ALPHA * xv.z;
        r.w = (1.0f - ALPHA) * a.w + ALPHA * xv.w;
        out4[i] = r;
        float4 z; z.x = 0.f; z.y = 0.f; z.z = 0.f; z.w = 0.f;
        agg4[i] = z;
    }
}

extern "C" void kernel_launch(void* const* d_in, const int* in_sizes, int n_in,
                              void* d_out, int out_size, void* d_ws, size_t ws_size,
                              hipStream_t stream)
{
    const float* x    = (const float*)d_in[0];   // [N, 64] f32
    const int*   erow = (const int*)  d_in[1];   // [E] i32
    const int*   ecol = (const int*)  d_in[2];   // [E] i32
    const float* ew   = (const float*)d_in[3];   // [E] f32

    const int E  = in_sizes[1];
    const int ND = out_size;                     // N * 64

    float* out = (float*)d_out;                  // h lives here across iterations
    float* agg = (float*)d_ws;                   // ND floats of scratch

    // zero the aggregation buffer (ws is poisoned before timing; combine()
    // re-zeros it every iteration so replayed calls also start clean).
    hipMemsetAsync(agg, 0, (size_t)ND * sizeof(float), stream);

    const int scatterBlocks = (E + EPB - 1) / EPB;
    const int n4            = ND / 4;
    const int combBlocks    = (n4 + 255) / 256;

    const float* src = x;                        // h0 = x
    for (int it = 0; it < K_STEPS; ++it) {
        appnp_scatter<<<scatterBlocks, 256, 0, stream>>>(src, erow, ecol, ew, agg, E);
        appnp_combine<<<combBlocks, 256, 0, stream>>>((const float4*)x,
                                                      (float4*)agg,
                                                      (float4*)out, n4);
        src = out;
    }
}